// SRLoss_67448166416691
// MI455X (gfx1250) — compile-verified
//
#include <hip/hip_runtime.h>
#include <hip/hip_bf16.h>

// ---------------------------------------------------------------------------
// SRLoss for MI455X (gfx1250, wave32).
// loss = mean((inp - avgpool10x10(output))^2) - mean(t*log(om) + (1-t)*log(oc))
// Fused single pass over `output` (BCE + pooling via LDS stripe), deterministic
// two-kernel tree reduction. Cross-lane reduction uses V_WMMA_F32_16X16X4_F32
// with a ones B-matrix (A row-sums land in every column of D).
// ---------------------------------------------------------------------------

typedef __attribute__((ext_vector_type(2))) float v2f;
typedef __attribute__((ext_vector_type(8))) float v8f;

#define SCALE 10
#define BATCH 32
#define HH 1280
#define WW 1280
#define PH (HH / SCALE)   // 128
#define PW (WW / SCALE)   // 128
#define STRIPE_ELEMS (SCALE * WW)     // 12800 floats per block stripe
#define NBLOCKS (BATCH * PH)          // 4096
#define NT1 320                       // 10 waves; 12800 = 320*4*10 (exact fl4)
#define NT2 256
#define NWAVES1 (NT1 / 32)            // 10

// Sum a value across all 32 lanes of the wave using one f32 WMMA.
// D = A(16x4) * ones(4x16): every column n of D holds the 16 row-sums of A,
// so per-lane sum of the 8 C/D VGPRs gives rows 0..7 (lanes 0-15) or rows
// 8..15 (lanes 16-31); xor-shuffle by 16 combines them. EXEC must be all 1s.
__device__ __forceinline__ float wave_sum32(float v) {
  v2f a; a.x = v;    a.y = 0.0f;
  v2f b; b.x = 1.0f; b.y = 1.0f;
  v8f c = {};
  c = __builtin_amdgcn_wmma_f32_16x16x4_f32(
      /*neg_a=*/false, a, /*neg_b=*/false, b,
      /*c_mod=*/(short)0, c, /*reuse_a=*/false, /*reuse_b=*/false);
  float s = c[0] + c[1] + c[2] + c[3] + c[4] + c[5] + c[6] + c[7];
  s += __shfl_xor(s, 16, 32);
  return s;
}

// t*log(om) + (1-t)*log(oc) = lc + t*(lm - lc).
// Inputs are multiples of ~2^-23 from jax.random.uniform, so any nonzero value
// is >= ~1.19e-7 >> 1e-20: fmaxf(x, eps) == where(x==0, eps, x) bit-exactly.
__device__ __forceinline__ float bce_elem(float o, float t) {
  const float kEps = 1e-20f;
  float lm = __logf(fmaxf(o, kEps));
  float lc = __logf(fmaxf(1.0f - o, kEps));
  return lc + t * (lm - lc);
}

__global__ __launch_bounds__(NT1)
void srloss_stripe_kernel(const float* __restrict__ output,
                          const float* __restrict__ target,
                          const float* __restrict__ inp,
                          float* __restrict__ partials) {
  __shared__ __align__(16) float s_stripe[STRIPE_ELEMS];  // 51.2 KB LDS
  __shared__ float s_wave[2 * NWAVES1];

  const int tid  = threadIdx.x;
  const int blk  = blockIdx.x;
  const int b    = blk >> 7;    // / PH
  const int rblk = blk & 127;   // % PH
  const size_t base = ((size_t)b * HH + (size_t)rblk * SCALE) * (size_t)WW;

  const float4* out4 = (const float4*)(output + base);
  const float4* tgt4 = (const float4*)(target + base);
  float4* s4 = (float4*)s_stripe;

  // Phase A: stream stripe with b128 loads, BCE on the fly, stage in LDS.
  float bce = 0.0f;
  #pragma unroll 5
  for (int k = 0; k < STRIPE_ELEMS / 4 / NT1; ++k) {  // 10 iterations
    const int i4 = k * NT1 + tid;
    const float4 o = out4[i4];
    const float4 t = tgt4[i4];
    s4[i4] = o;
    bce += bce_elem(o.x, t.x);
    bce += bce_elem(o.y, t.y);
    bce += bce_elem(o.z, t.z);
    bce += bce_elem(o.w, t.w);
  }
  __syncthreads();

  // Phase B: 128 threads each pool one 10x10 tile from LDS (b64 reads;
  // {10j,10j+1} mod 64 bank pairs are disjoint across 32 lanes -> no conflict).
  float mse = 0.0f;
  if (tid < PW) {
    float ps = 0.0f;
    #pragma unroll
    for (int r = 0; r < SCALE; ++r) {
      const float2* row2 = (const float2*)(s_stripe + r * WW + tid * SCALE);
      #pragma unroll
      for (int c = 0; c < SCALE / 2; ++c) {
        const float2 v = row2[c];
        ps += v.x + v.y;
      }
    }
    const float pooled = ps * (1.0f / (SCALE * SCALE));
    const float d = inp[((size_t)b * PH + rblk) * PW + tid] - pooled;
    mse = d * d;
  }

  // Block reduction: every wave runs WMMA with full EXEC, leaders -> LDS.
  const float sb = wave_sum32(bce);
  const float sm = wave_sum32(mse);
  const int wave = tid >> 5, lane = tid & 31;
  if (lane == 0) { s_wave[wave] = sb; s_wave[NWAVES1 + wave] = sm; }
  __syncthreads();
  if (tid == 0) {
    float tb = 0.0f, tm = 0.0f;
    #pragma unroll
    for (int w = 0; w < NWAVES1; ++w) { tb += s_wave[w]; tm += s_wave[NWAVES1 + w]; }
    partials[2 * blk]     = tb;
    partials[2 * blk + 1] = tm;
  }
}

__global__ __launch_bounds__(NT2)
void srloss_finalize_kernel(const float* __restrict__ partials,
                            float* __restrict__ out) {
  __shared__ float s_wave[16];
  const int tid = threadIdx.x;

  float tb = 0.0f, tm = 0.0f;
  for (int i = tid; i < NBLOCKS; i += NT2) {  // fixed order: deterministic
    tb += partials[2 * i];
    tm += partials[2 * i + 1];
  }

  const float sb = wave_sum32(tb);
  const float sm = wave_sum32(tm);
  const int wave = tid >> 5, lane = tid & 31;
  if (lane == 0) { s_wave[wave] = sb; s_wave[8 + wave] = sm; }
  __syncthreads();
  if (tid == 0) {
    float Bt = 0.0f, Mt = 0.0f;
    #pragma unroll
    for (int w = 0; w < 8; ++w) { Bt += s_wave[w]; Mt += s_wave[8 + w]; }
    const float n_bce = (float)((size_t)BATCH * HH * WW);      // 52428800
    const float n_mse = (float)((size_t)BATCH * PH * PW);      // 524288
    out[0] = Mt / n_mse - Bt / n_bce;   // mse + (-mean(bce_sum))
  }
}

extern "C" void kernel_launch(void* const* d_in, const int* in_sizes, int n_in,
                              void* d_out, int out_size, void* d_ws, size_t ws_size,
                              hipStream_t stream) {
  const float* output = (const float*)d_in[0];  // [32,1,1280,1280]
  const float* target = (const float*)d_in[1];  // [32,1,1280,1280]
  const float* inp    = (const float*)d_in[2];  // [32,1,128,128]
  float* out      = (float*)d_out;              // 1 element
  float* partials = (float*)d_ws;               // 2 * 4096 floats = 32 KB

  srloss_stripe_kernel<<<NBLOCKS, NT1, 0, stream>>>(output, target, inp, partials);
  srloss_finalize_kernel<<<1, NT2, 0, stream>>>(partials, out);
}